// AGNN_DGL_67156108640390
// MI455X (gfx1250) — compile-verified
//
#include <hip/hip_runtime.h>
#include <math.h>

#define N_NODES 65536
#define N_EDGES 1048576
#define IN_FEATS 128
#define N_HIDDEN 64

typedef __attribute__((ext_vector_type(2))) float v2f;
typedef __attribute__((ext_vector_type(8))) float v8f;

// ---------------------------------------------------------------------------
// GEMM: out[N_NODES,64] = A[N_NODES,K] @ W[K,64] + bias  (optional ReLU)
// Native FP32 WMMA: V_WMMA_F32_16X16X4_F32.
// Block = 256 threads = 8 waves; each wave owns a 16-row x 64-col tile.
//
// W is staged in LDS in K-pair-interleaved, bank-swizzled form:
//   W[k][col] -> wlds[(k>>1)*128 + ((col ^ (((k>>1)&1)<<4)) << 1) + (k&1)]
// so each B fragment (W[kk][col], W[kk+1][col]) is a single aligned
// ds_load_b64 into an even VGPR pair (no reassembly movs), and the XOR-16
// swizzle gives the two half-waves (reading rows k2 and k2+1) disjoint
// LDS bank sets.
//
// A fragment (16x4 f32): lanes 0-15 -> M=lane, K=k..k+1; lanes 16-31 -> K=k+2..k+3
// C/D (16x16 f32): VGPR v, lane L -> M = v + 8*(L>=16), N = L%16
// ---------------------------------------------------------------------------
template <int K, bool RELU>
__global__ __launch_bounds__(256) void gemm_bias_kernel(
    const float* __restrict__ A, const float* __restrict__ W,
    const float* __restrict__ bias, float* __restrict__ out) {
  __shared__ float wlds[K * 64];
  const int tid = threadIdx.x;
  for (int i = tid; i < K * 64; i += 256) {
    const int k = i >> 6;
    const int col = i & 63;
    const int k2 = k >> 1;
    wlds[k2 * 128 + (((col ^ ((k2 & 1) << 4))) << 1) + (k & 1)] = W[i];
  }
  __syncthreads();

  const int wave = tid >> 5;
  const int lane = tid & 31;
  const int l16 = lane & 15;
  const int lh = lane >> 4;  // 0 or 1
  const int rowBase = (blockIdx.x * 8 + wave) * 16;

  v8f acc[4] = {v8f{0}, v8f{0}, v8f{0}, v8f{0}};
  const float* arow = A + (size_t)(rowBase + l16) * K;
  // per-lane LDS base: row (k2 = k/2 + lh), column swizzled by lh
  const float* bbase = wlds + lh * 128 + ((l16 ^ (lh << 4)) << 1);

  for (int k = 0; k < K; k += 4) {
    v2f a = *(const v2f*)(arow + k + 2 * lh);  // 8B aligned
    const float* bk = bbase + (k >> 1) * 128;
#pragma unroll
    for (int nt = 0; nt < 4; ++nt) {
      v2f b = *(const v2f*)(bk + nt * 32);  // ds_load_b64, offsets 0/128/256/384B
      acc[nt] = __builtin_amdgcn_wmma_f32_16x16x4_f32(
          false, a, false, b, (short)0, acc[nt], false, false);
    }
  }

#pragma unroll
  for (int nt = 0; nt < 4; ++nt) {
    const int col = nt * 16 + l16;
    const float bv = bias[col];
#pragma unroll
    for (int v = 0; v < 8; ++v) {
      const int row = rowBase + v + 8 * lh;
      float x = acc[nt][v] + bv;
      if (RELU) x = fmaxf(x, 0.0f);
      out[(size_t)row * 64 + col] = x;
    }
  }
}

// ---------------------------------------------------------------------------
// Row L2-normalize: nh = h / max(||h||_2, eps).  Wave per node, float2/lane.
// ---------------------------------------------------------------------------
__global__ __launch_bounds__(256) void l2norm_kernel(const float* __restrict__ h,
                                                     float* __restrict__ nh) {
  const int wave = threadIdx.x >> 5, lane = threadIdx.x & 31;
  const int node = blockIdx.x * 8 + wave;
  const size_t base = (size_t)node * 64 + lane * 2;
  float2 v = *(const float2*)(h + base);
  float ss = v.x * v.x + v.y * v.y;
#pragma unroll
  for (int m = 16; m >= 1; m >>= 1) ss += __shfl_xor(ss, m, 32);
  const float inv = 1.0f / fmaxf(sqrtf(ss), 1e-12f);
  float2 o;
  o.x = v.x * inv;
  o.y = v.y * inv;
  *(float2*)(nh + base) = o;
}

// ---------------------------------------------------------------------------
// Per-layer init: zero aggregation target, m = -inf, s = 0.
// ---------------------------------------------------------------------------
__global__ void init_kernel(float* __restrict__ hNext, float* __restrict__ m,
                            float* __restrict__ s) {
  const int i = blockIdx.x * blockDim.x + threadIdx.x;  // over N_NODES*64
  hNext[i] = 0.0f;
  if (i < N_NODES) {
    m[i] = -__builtin_inff();
    s[i] = 0.0f;
  }
}

// ---------------------------------------------------------------------------
// Edge logits: e = beta * <nh[src], nh[dst]>, plus segment-max via HW f32
// atomic max. Wave per edge; gathers are 256B coalesced wave loads that
// hit L2 (nh is 16MB << 192MB L2).
// ---------------------------------------------------------------------------
__global__ __launch_bounds__(256) void edge_dot_kernel(
    const float* __restrict__ nh, const int* __restrict__ src,
    const int* __restrict__ dst, const float* __restrict__ betas, int betaIdx,
    float* __restrict__ ev, float* __restrict__ m) {
  const int wave = threadIdx.x >> 5, lane = threadIdx.x & 31;
  const int e = blockIdx.x * 8 + wave;
  const int sN = src[e], dN = dst[e];
  float2 a = *(const float2*)(nh + (size_t)sN * 64 + lane * 2);
  float2 b = *(const float2*)(nh + (size_t)dN * 64 + lane * 2);
  float p = a.x * b.x + a.y * b.y;
#pragma unroll
  for (int mm = 16; mm >= 1; mm >>= 1) p += __shfl_xor(p, mm, 32);
  if (lane == 0) {
    const float val = betas[betaIdx] * p;
    ev[e] = val;
    unsafeAtomicMax(m + dN, val);  // global_atomic_max_num_f32
  }
}

// ---------------------------------------------------------------------------
// ex = exp(e - m[dst]); s[dst] += ex.  Thread per edge.
// ---------------------------------------------------------------------------
__global__ void expsum_kernel(float* __restrict__ ev, const int* __restrict__ dst,
                              const float* __restrict__ m, float* __restrict__ s) {
  const int e = blockIdx.x * blockDim.x + threadIdx.x;
  const int d = dst[e];
  const float x = __expf(ev[e] - m[d]);
  ev[e] = x;
  unsafeAtomicAdd(s + d, x);  // global_atomic_add_f32
}

// ---------------------------------------------------------------------------
// hNext[dst] += (ex/s[dst]) * h[src].  Wave per edge, 2 f32 atomics per lane.
// ---------------------------------------------------------------------------
__global__ __launch_bounds__(256) void agg_kernel(
    const float* __restrict__ h, const int* __restrict__ src,
    const int* __restrict__ dst, const float* __restrict__ ev,
    const float* __restrict__ s, float* __restrict__ hNext) {
  const int wave = threadIdx.x >> 5, lane = threadIdx.x & 31;
  const int e = blockIdx.x * 8 + wave;
  const int sN = src[e], dN = dst[e];
  const float alpha = ev[e] / s[dN];
  float2 hv = *(const float2*)(h + (size_t)sN * 64 + lane * 2);
  float* o = hNext + (size_t)dN * 64 + lane * 2;
  unsafeAtomicAdd(o, alpha * hv.x);
  unsafeAtomicAdd(o + 1, alpha * hv.y);
}

// ---------------------------------------------------------------------------
extern "C" void kernel_launch(void* const* d_in, const int* in_sizes, int n_in,
                              void* d_out, int out_size, void* d_ws,
                              size_t ws_size, hipStream_t stream) {
  const float* features = (const float*)d_in[0];
  const int* src = (const int*)d_in[1];
  const int* dst = (const int*)d_in[2];
  const float* W1 = (const float*)d_in[3];
  const float* b1 = (const float*)d_in[4];
  const float* W2 = (const float*)d_in[5];
  const float* b2 = (const float*)d_in[6];
  const float* betas = (const float*)d_in[7];
  float* out = (float*)d_out;

  char* ws = (char*)d_ws;
  const size_t HBYTES = (size_t)N_NODES * 64 * sizeof(float);  // 16 MB
  float* hA = (float*)(ws);
  float* hB = (float*)(ws + HBYTES);
  float* nh = (float*)(ws + 2 * HBYTES);
  float* ev = (float*)(ws + 3 * HBYTES);
  float* m = (float*)(ws + 3 * HBYTES + (size_t)N_EDGES * sizeof(float));
  float* s = m + N_NODES;

  // proj: h = relu(features @ W1 + b1)
  gemm_bias_kernel<IN_FEATS, true><<<N_NODES / 128, 256, 0, stream>>>(
      features, W1, b1, hA);

  float* hin = hA;
  float* hout = hB;
  for (int layer = 0; layer < 2; ++layer) {
    l2norm_kernel<<<N_NODES / 8, 256, 0, stream>>>(hin, nh);
    init_kernel<<<(N_NODES * 64) / 256, 256, 0, stream>>>(hout, m, s);
    edge_dot_kernel<<<N_EDGES / 8, 256, 0, stream>>>(nh, src, dst, betas,
                                                     layer, ev, m);
    expsum_kernel<<<N_EDGES / 256, 256, 0, stream>>>(ev, dst, m, s);
    agg_kernel<<<N_EDGES / 8, 256, 0, stream>>>(hin, src, dst, ev, s, hout);
    float* t = hin;
    hin = hout;
    hout = t;
  }

  // cls: out = h @ W2 + b2
  gemm_bias_kernel<N_HIDDEN, false><<<N_NODES / 128, 256, 0, stream>>>(
      hin, W2, b2, out);
}